// MambaCore_90039694393636
// MI455X (gfx1250) — compile-verified
//
#include <hip/hip_runtime.h>
#include <hip/hip_bf16.h>
#include <math.h>

// ---------------- problem constants (match reference) ----------------
#define D_MODEL   1024
#define D_STATE   16
#define D_CONVW   4
#define D_INNER   2048           // EXPAND * D_MODEL
#define DT_RANK   64             // (D_MODEL + 15)/16
#define BATCH     2
#define SEQ_LEN   2048
#define NROW      (BATCH * SEQ_LEN)          // 4096 flattened (b,l) rows
#define XPROJ_N   (DT_RANK + 2 * D_STATE)    // 96

typedef __attribute__((ext_vector_type(2))) float v2f;
typedef __attribute__((ext_vector_type(8))) float v8f;

// =====================================================================
// NT-GEMM with fp32 WMMA:  C[M,N] = A[M,K] * W[N,K]^T   (row-major all)
// One wave computes a (16*MT) x (16*NT) tile; K stepped by 4 per
// v_wmma_f32_16x16x4_f32.  Fragment loads are double-buffered one K-step
// ahead of the WMMAs so the matrix pipe is not gated on fresh loads.
// K is a template constant (trip counts fold; guards disappear).
// EPI==1 fuses the dt epilogue: c = min(softplus(c + bias[n]), 10).
// Grid must supply exactly (M/(16*MT))*(N/(16*NT)) waves. K % 8 == 0.
// =====================================================================
template <int MT, int NT, int K, int EPI>
__global__ __launch_bounds__(256) void gemm_nt_f32wmma(
    const float* __restrict__ A, int lda,
    const float* __restrict__ W, int ldw,
    float* __restrict__ C, int ldc,
    int M, int N,
    const float* __restrict__ bias)
{
    const int lane = threadIdx.x & 31;
    const int wave = (blockIdx.x * blockDim.x + threadIdx.x) >> 5;
    const int ngrp = N / (16 * NT);
    const int m0 = (wave / ngrp) * (16 * MT);
    const int n0 = (wave % ngrp) * (16 * NT);
    if (m0 >= M) return;                       // wave-uniform; EXEC stays all-1s

    const int l16  = lane & 15;
    const int koff = (lane >> 4) << 1;         // 0 for lanes 0-15, 2 for 16-31

    v8f zero = {0.f, 0.f, 0.f, 0.f, 0.f, 0.f, 0.f, 0.f};
    v8f acc[MT][NT];
#pragma unroll
    for (int i = 0; i < MT; ++i)
#pragma unroll
        for (int j = 0; j < NT; ++j) acc[i][j] = zero;

    // A fragment (16x4 f32 layout): lane m = lane&15, VGPR0/1 hold K=koff,koff+1
    // B fragment (4x16 f32 layout): lane n = lane&15, K=koff,koff+1 (B[k][n]==W[n][k])
    const float* Abase = A + (size_t)(m0 + l16) * lda + koff;
    const float* Wbase = W + (size_t)(n0 + l16) * ldw + koff;

    v2f af[2][MT], bf[2][NT];

    auto loadfrag = [&](int buf, int k) {
#pragma unroll
        for (int i = 0; i < MT; ++i)
            af[buf][i] = *(const v2f*)(Abase + (size_t)i * 16 * lda + k);
#pragma unroll
        for (int j = 0; j < NT; ++j)
            bf[buf][j] = *(const v2f*)(Wbase + (size_t)j * 16 * ldw + k);
    };
    auto compute = [&](int buf) {
#pragma unroll
        for (int i = 0; i < MT; ++i)
#pragma unroll
            for (int j = 0; j < NT; ++j)
                acc[i][j] = __builtin_amdgcn_wmma_f32_16x16x4_f32(
                    false, af[buf][i], false, bf[buf][j],
                    (short)0, acc[i][j], false, false);
    };

    // Software pipeline, depth 1 (loads one K-step ahead of consumption).
    loadfrag(0, 0);
    for (int k0 = 0; k0 < K; k0 += 8) {
        loadfrag(1, k0 + 4);                   // always in-bounds: K % 8 == 0
        compute(0);
        if (k0 + 8 < K) loadfrag(0, k0 + 8);
        compute(1);
    }

    // C/D 16x16 f32 layout: VGPR v -> row +v (lanes 0-15), +8+v (lanes 16-31).
#pragma unroll
    for (int i = 0; i < MT; ++i) {
        const int mbase = m0 + i * 16 + ((lane >> 4) << 3);
#pragma unroll
        for (int j = 0; j < NT; ++j) {
            const int n = n0 + j * 16 + l16;
#pragma unroll
            for (int v = 0; v < 8; ++v) {
                float val = acc[i][j][v];
                if (EPI == 1) {
                    val += bias[n];
                    val = fminf(logf(1.f + __expf(val)), 10.f);  // softplus,clamp
                }
                C[(size_t)(mbase + v) * ldc + n] = val;
            }
        }
    }
}

// =====================================================================
// Depthwise causal conv (width 4) + bias + SiLU.
// xr is [NROW, 2*D_INNER]; channel half [0, D_INNER) is xs.
// u out is [NROW, D_INNER].
// =====================================================================
__global__ __launch_bounds__(256) void conv_silu_kernel(
    const float* __restrict__ xr,
    const float* __restrict__ cw,   // [D_INNER, 1, 4]
    const float* __restrict__ cb,   // [D_INNER]
    float* __restrict__ u)
{
    const int idx = blockIdx.x * blockDim.x + threadIdx.x;  // (row, d)
    const int d   = idx & (D_INNER - 1);
    const int row = idx >> 11;                              // b*L + l
    const int l   = row & (SEQ_LEN - 1);

    float acc = cb[d];
#pragma unroll
    for (int j = 0; j < D_CONVW; ++j) {
        const int ls = l - (D_CONVW - 1) + j;               // causal pad left
        if (ls >= 0)
            acc += xr[(size_t)(row - (D_CONVW - 1) + j) * (2 * D_INNER) + d] *
                   cw[d * D_CONVW + j];
    }
    u[idx] = acc / (1.f + __expf(-acc));                    // SiLU
}

// =====================================================================
// Selective scan, fused with res-gate.
// Lane layout: 16 lanes per channel (one per state s), 2 channels/wave.
// Sequential over t; 16-wide xor-shuffle tree reduces h*C each step.
// yg[row,d] = (sum_s h*C + D*u) * silu(res)
// =====================================================================
__global__ __launch_bounds__(256) void mamba_scan_kernel(
    const float* __restrict__ dt,     // [NROW, D_INNER]
    const float* __restrict__ u,      // [NROW, D_INNER]
    const float* __restrict__ xdbl,   // [NROW, 96] (B at +64, C at +80)
    const float* __restrict__ xr,     // [NROW, 2*D_INNER] (res at +D_INNER)
    const float* __restrict__ A_log,  // [D_INNER, 16]
    const float* __restrict__ Dvec,   // [D_INNER]
    float* __restrict__ yg)           // [NROW, D_INNER]
{
    const int lane = threadIdx.x & 31;
    const int wave = (blockIdx.x * blockDim.x + threadIdx.x) >> 5;
    const int s  = lane & 15;
    const int ch = wave * 2 + (lane >> 4);    // global channel in [0, B*D_INNER)
    const int b  = ch / D_INNER;
    const int d  = ch & (D_INNER - 1);

    const float As = -__expf(A_log[d * D_STATE + s]);  // A = -exp(A_log)
    const float Dd = Dvec[d];

    float h = 0.f;
    const int rowbase = b * SEQ_LEN;
    for (int t = 0; t < SEQ_LEN; ++t) {
        const int r = rowbase + t;
        const float dtv = dt[(size_t)r * D_INNER + d];
        const float uv  = u[(size_t)r * D_INNER + d];
        const float Bv  = xdbl[r * XPROJ_N + DT_RANK + s];
        const float Cv  = xdbl[r * XPROJ_N + DT_RANK + D_STATE + s];

        const float dA = __expf(dtv * As);
        h = dA * h + (dtv * uv) * Bv;

        float p = h * Cv;
        p += __shfl_xor(p, 1, 16);
        p += __shfl_xor(p, 2, 16);
        p += __shfl_xor(p, 4, 16);
        p += __shfl_xor(p, 8, 16);

        if (s == 0) {
            const float yv = p + Dd * uv;
            const float rv = xr[(size_t)r * (2 * D_INNER) + D_INNER + d];
            const float g  = rv / (1.f + __expf(-rv));  // silu(res)
            yg[(size_t)r * D_INNER + d] = yv * g;
        }
    }
}

// =====================================================================
// Host-side orchestration
// =====================================================================
extern "C" void kernel_launch(void* const* d_in, const int* in_sizes, int n_in,
                              void* d_out, int out_size, void* d_ws, size_t ws_size,
                              hipStream_t stream)
{
    const float* x        = (const float*)d_in[0];  // [B, L, D_MODEL]
    const float* in_w     = (const float*)d_in[1];  // [2*D_INNER, D_MODEL]
    const float* conv_w   = (const float*)d_in[2];  // [D_INNER, 1, 4]
    const float* conv_b   = (const float*)d_in[3];  // [D_INNER]
    const float* xproj_w  = (const float*)d_in[4];  // [96, D_INNER]
    const float* dtproj_w = (const float*)d_in[5];  // [D_INNER, DT_RANK]
    const float* dtproj_b = (const float*)d_in[6];  // [D_INNER]
    const float* A_log    = (const float*)d_in[7];  // [D_INNER, 16]
    const float* Dvec     = (const float*)d_in[8];  // [D_INNER]
    const float* out_w    = (const float*)d_in[9];  // [D_MODEL, D_INNER]
    float* out = (float*)d_out;                     // [B, L, D_MODEL]

    // Workspace carve-up (fp32), 256B aligned by construction.
    char* ws = (char*)d_ws;
    float* xr   = (float*)ws;                                   // [4096, 4096]  64 MB
    float* u    = (float*)(ws + (size_t)NROW * 4096 * 4);       // [4096, 2048]  32 MB
    float* xdbl = (float*)(ws + (size_t)NROW * 6144 * 4);       // [4096,   96] 1.5 MB
    float* dt   = (float*)(ws + (size_t)NROW * 6240 * 4);       // [4096, 2048]  32 MB
    float* yg   = (float*)(ws + (size_t)NROW * 8288 * 4);       // [4096, 2048]  32 MB
    // total = NROW * 10336 * 4 B  ~= 162 MB

    // 1) in_proj: xr[4096,4096] = x[4096,1024] @ in_w[4096,1024]^T
    //    waves = (4096/32)*(4096/64) = 8192 -> 1024 blocks of 8 waves
    gemm_nt_f32wmma<2, 4, D_MODEL, 0><<<1024, 256, 0, stream>>>(
        x, D_MODEL, in_w, D_MODEL, xr, 2 * D_INNER,
        NROW, 2 * D_INNER, nullptr);

    // 2) depthwise conv + SiLU -> u[4096,2048]
    conv_silu_kernel<<<(NROW * D_INNER) / 256, 256, 0, stream>>>(
        xr, conv_w, conv_b, u);

    // 3) x_proj: xdbl[4096,96] = u @ xproj_w[96,2048]^T
    //    waves = (4096/32)*(96/32) = 384 -> 48 blocks
    gemm_nt_f32wmma<2, 2, D_INNER, 0><<<48, 256, 0, stream>>>(
        u, D_INNER, xproj_w, D_INNER, xdbl, XPROJ_N,
        NROW, XPROJ_N, nullptr);

    // 4) dt_proj + bias + softplus + clamp:
    //    dt[4096,2048] = min(softplus(xdbl[:, :64] @ dtproj_w^T + b), 10)
    //    waves = (4096/32)*(2048/64) = 4096 -> 512 blocks
    gemm_nt_f32wmma<2, 4, DT_RANK, 1><<<512, 256, 0, stream>>>(
        xdbl, XPROJ_N, dtproj_w, DT_RANK, dt, D_INNER,
        NROW, D_INNER, dtproj_b);

    // 5) selective scan + D*u + silu(res) gate -> yg[4096,2048]
    //    channels = B*D_INNER = 4096 -> 2048 waves -> 256 blocks
    mamba_scan_kernel<<<256, 256, 0, stream>>>(
        dt, u, xdbl, xr, A_log, Dvec, yg);

    // 6) out_proj: out[4096,1024] = yg @ out_w[1024,2048]^T
    //    waves = (4096/32)*(1024/64) = 2048 -> 256 blocks
    gemm_nt_f32wmma<2, 4, D_INNER, 0><<<256, 256, 0, stream>>>(
        yg, D_INNER, out_w, D_INNER, out, D_MODEL,
        NROW, D_MODEL, nullptr);
}